// AFNO2DLayer_11751030521904
// MI455X (gfx1250) — compile-verified
//
#include <hip/hip_runtime.h>
#include <cstddef>

#define TWO_PI 6.283185307179586f

typedef float v2f __attribute__((ext_vector_type(2)));
typedef float v8f __attribute__((ext_vector_type(8)));

// ---------------------------------------------------------------------------
// f32 WMMA: D(16x16) = A(16x4) * B(4x16) + C  — V_WMMA_F32_16X16X4_F32
// ---------------------------------------------------------------------------
__device__ __forceinline__ v8f wmma4(v2f a, v2f b, v8f c) {
  return __builtin_amdgcn_wmma_f32_16x16x4_f32(
      /*neg_a=*/false, a, /*neg_b=*/false, b,
      /*c_mod=*/(short)0, c, /*reuse_a=*/false, /*reuse_b=*/false);
}

// A fragment: row-major A[16][lda], rows M=lane&15; lanes 16-31 hold K+2,K+3
__device__ __forceinline__ v2f fragA(const float* A, int lda, int k0, int lane) {
  const float* p = A + (lane & 15) * lda + k0 + ((lane >> 4) << 1);
  v2f a;
  a.x = p[0];
  a.y = p[1];
  return a;
}

// B fragment: row-major B[K][ldb], cols N=lane&15; lanes 16-31 hold K+2,K+3
__device__ __forceinline__ v2f fragB(const float* B, int ldb, int col0, int k0, int lane) {
  const float* p = B + (k0 + ((lane >> 4) << 1)) * ldb + col0 + (lane & 15);
  v2f b;
  b.x = p[0];
  b.y = p[ldb];
  return b;
}

// ---------------------------------------------------------------------------
// Stage 1: rDFT along W.  Per (b,h): out[wf(65),c] = F_w[wf,w] * x[w,c]
// ortho norm 1/sqrt(H*W)=1/128 folded into coefficients.
// grid (6 Ntiles, 5 Mtiles, 512 bh), block 256 (8 waves)
// ---------------------------------------------------------------------------
__global__ void wdft_kernel(const float* __restrict__ x,
                            float* __restrict__ outR, float* __restrict__ outI) {
  extern __shared__ float lds[];
  float* Ar = lds;              // 16*128
  float* Ai = Ar + 16 * 128;    // 16*128
  float* Bx = Ai + 16 * 128;    // 128*128
  const int tid = threadIdx.x;
  const int bh  = blockIdx.z;          // b*128 + h
  const int wf0 = blockIdx.y * 16;
  const int c0  = blockIdx.x * 128;
  const float* xrow = x + (size_t)bh * 128 * 768;
  // vectorized b128 staging: 128 rows x 32 float4
  float4* Bx4 = reinterpret_cast<float4*>(Bx);
  for (int idx = tid; idx < 128 * 32; idx += 256) {
    int w = idx >> 5, c4 = idx & 31;
    Bx4[idx] = reinterpret_cast<const float4*>(xrow + (size_t)w * 768 + c0)[c4];
  }
  for (int idx = tid; idx < 16 * 128; idx += 256) {
    int m = idx >> 7, w = idx & 127;
    int p = ((wf0 + m) * w) & 127;     // phase mod 128
    float s, c;
    __sincosf(-(TWO_PI / 128.f) * (float)p, &s, &c);
    Ar[idx] = c * (1.f / 128.f);       // Re exp(-i t)
    Ai[idx] = s * (1.f / 128.f);       // Im exp(-i t)
  }
  __syncthreads();
  const int lane = tid & 31, wv = tid >> 5;
  const int cs = wv * 16;
  v8f accR = {}, accI = {};
  for (int k0 = 0; k0 < 128; k0 += 4) {
    v2f b = fragB(Bx, 128, cs, k0, lane);
    accR = wmma4(fragA(Ar, 128, k0, lane), b, accR);
    accI = wmma4(fragA(Ai, 128, k0, lane), b, accI);
  }
  const int nn = lane & 15, hi = lane >> 4;
  for (int i = 0; i < 8; ++i) {
    int wf = wf0 + i + (hi << 3);
    if (wf < 65) {
      size_t off = ((size_t)bh * 65 + wf) * 768 + c0 + cs + nn;
      outR[off] = accR[i];
      outI[off] = accI[i];
    }
  }
}

// ---------------------------------------------------------------------------
// Stages 2 & 4: complex DFT along H. Per (b,wf):
// out[k,c] = exp(sign*i*2pi*k*h/128)[k,h] * in[h,c]   (complex GEMM, 4 real)
// grid (6 Ntiles, 8 Mtiles, 260 b*wf), block 256
// ---------------------------------------------------------------------------
__global__ void hdft_kernel(const float* __restrict__ inR, const float* __restrict__ inI,
                            float* __restrict__ outR, float* __restrict__ outI,
                            float sign) {
  extern __shared__ float lds[];
  float* Ar = lds;               // 16*128
  float* Ai = Ar + 16 * 128;     // 16*128
  float* Br = Ai + 16 * 128;     // 128*128
  float* Bi = Br + 128 * 128;    // 128*128
  const int tid = threadIdx.x;
  const int bwf = blockIdx.z;    // b*65 + wf
  const int b = bwf / 65, wf = bwf % 65;
  const int k0row = blockIdx.y * 16;
  const int c0 = blockIdx.x * 128;
  float4* Br4 = reinterpret_cast<float4*>(Br);
  float4* Bi4 = reinterpret_cast<float4*>(Bi);
  for (int idx = tid; idx < 128 * 32; idx += 256) {
    int h = idx >> 5, c4 = idx & 31;
    size_t off = (((size_t)b * 128 + h) * 65 + wf) * 768 + c0;   // float index, 16B aligned
    Br4[idx] = reinterpret_cast<const float4*>(inR + off)[c4];
    Bi4[idx] = reinterpret_cast<const float4*>(inI + off)[c4];
  }
  for (int idx = tid; idx < 16 * 128; idx += 256) {
    int m = idx >> 7, h = idx & 127;
    int p = ((k0row + m) * h) & 127;
    float s, c;
    __sincosf(sign * (TWO_PI / 128.f) * (float)p, &s, &c);
    Ar[idx] = c;
    Ai[idx] = s;
  }
  __syncthreads();
  const int lane = tid & 31, wv = tid >> 5;
  const int cs = wv * 16;
  v8f aRR = {}, aII = {}, aRI = {}, aIR = {};
  for (int k0 = 0; k0 < 128; k0 += 4) {
    v2f ar = fragA(Ar, 128, k0, lane);
    v2f ai = fragA(Ai, 128, k0, lane);
    v2f br = fragB(Br, 128, cs, k0, lane);
    v2f bi = fragB(Bi, 128, cs, k0, lane);
    aRR = wmma4(ar, br, aRR);
    aII = wmma4(ai, bi, aII);
    aRI = wmma4(ar, bi, aRI);
    aIR = wmma4(ai, br, aIR);
  }
  const int nn = lane & 15, hi = lane >> 4;
  for (int i = 0; i < 8; ++i) {
    int k = k0row + i + (hi << 3);
    size_t off = (((size_t)b * 128 + k) * 65 + wf) * 768 + c0 + cs + nn;
    outR[off] = aRR[i] - aII[i];   // (Fr+iFi)(Xr+iXi): real
    outI[off] = aRI[i] + aIR[i];   // imag
  }
}

// ---------------------------------------------------------------------------
// Stage 3: block-diagonal complex MLP, fused 2 layers, in place.
// Rows r = (b*128+k)*65+wf (33280 total); per block n cols = n*96..n*96+95.
// grid (8 blocks, 260 row-tiles of 128), block 256 (8 waves)
// ---------------------------------------------------------------------------
__global__ void mlp_kernel(float* __restrict__ R, float* __restrict__ I,
                           const float* __restrict__ w1, const float* __restrict__ b1,
                           const float* __restrict__ w2, const float* __restrict__ b2) {
  extern __shared__ float lds[];
  float* Xr  = lds;               // 128*96
  float* Xi  = Xr + 128 * 96;
  float* Wr  = Xi + 128 * 96;     // 96*96
  float* Wi  = Wr + 96 * 96;
  float* O1r = Wi + 96 * 96;      // 128*96
  float* O1i = O1r + 128 * 96;
  float* bre = O1i + 128 * 96;    // 96
  float* bim = bre + 96;          // 96
  const int tid = threadIdx.x;    // 256
  const int n = blockIdx.x;
  const size_t row0 = (size_t)blockIdx.y * 128;

  const float* w1r = w1 + (size_t)n * 96 * 96;
  const float* w1i = w1 + (size_t)(8 + n) * 96 * 96;
  const float* w2r = w2 + (size_t)n * 96 * 96;
  const float* w2i = w2 + (size_t)(8 + n) * 96 * 96;

  // warm L2/L0 with layer-2 weights while layer 1 runs (global_prefetch_b8)
  if (tid < 144) {
    __builtin_prefetch(w2r + tid * 64, 0, 1);
    __builtin_prefetch(w2i + tid * 64, 0, 1);
  }

  // stage X tile (b128): 128 rows x 24 float4
  for (int idx = tid; idx < 128 * 24; idx += 256) {
    int r = idx / 24, c4 = idx % 24;
    size_t off4 = (((row0 + r) * 768 + n * 96) >> 2) + c4;
    reinterpret_cast<float4*>(Xr)[idx] = reinterpret_cast<const float4*>(R)[off4];
    reinterpret_cast<float4*>(Xi)[idx] = reinterpret_cast<const float4*>(I)[off4];
  }
  // stage W1/b1 (b128)
  for (int idx = tid; idx < 2304; idx += 256) {
    reinterpret_cast<float4*>(Wr)[idx] = reinterpret_cast<const float4*>(w1r)[idx];
    reinterpret_cast<float4*>(Wi)[idx] = reinterpret_cast<const float4*>(w1i)[idx];
  }
  if (tid < 24) {
    reinterpret_cast<float4*>(bre)[tid] = reinterpret_cast<const float4*>(b1 + n * 96)[tid];
    reinterpret_cast<float4*>(bim)[tid] = reinterpret_cast<const float4*>(b1 + (8 + n) * 96)[tid];
  }
  __syncthreads();

  const int lane = tid & 31, wv = tid >> 5;
  const int m0 = wv * 16;
  const int nn = lane & 15, hi = lane >> 4;

  // ----- layer 1: O1 = relu(X * W1 + b1)  (complex) -----
  for (int sub = 0; sub < 6; ++sub) {
    v8f aRR = {}, aII = {}, aRI = {}, aIR = {};
    for (int k0 = 0; k0 < 96; k0 += 4) {
      v2f ar  = fragA(Xr + m0 * 96, 96, k0, lane);
      v2f ai  = fragA(Xi + m0 * 96, 96, k0, lane);
      v2f brf = fragB(Wr, 96, sub * 16, k0, lane);
      v2f bif = fragB(Wi, 96, sub * 16, k0, lane);
      aRR = wmma4(ar, brf, aRR);
      aII = wmma4(ai, bif, aII);
      aRI = wmma4(ar, bif, aRI);
      aIR = wmma4(ai, brf, aIR);
    }
    int col = sub * 16 + nn;
    float cbr = bre[col], cbi = bim[col];
    for (int i = 0; i < 8; ++i) {
      int m = m0 + i + (hi << 3);
      float vr = aRR[i] - aII[i] + cbr;
      float vi = aRI[i] + aIR[i] + cbi;
      O1r[m * 96 + col] = vr > 0.f ? vr : 0.f;
      O1i[m * 96 + col] = vi > 0.f ? vi : 0.f;
    }
  }
  __syncthreads();
  // swap W2/b2 into W1's LDS region
  for (int idx = tid; idx < 2304; idx += 256) {
    reinterpret_cast<float4*>(Wr)[idx] = reinterpret_cast<const float4*>(w2r)[idx];
    reinterpret_cast<float4*>(Wi)[idx] = reinterpret_cast<const float4*>(w2i)[idx];
  }
  if (tid < 24) {
    reinterpret_cast<float4*>(bre)[tid] = reinterpret_cast<const float4*>(b2 + n * 96)[tid];
    reinterpret_cast<float4*>(bim)[tid] = reinterpret_cast<const float4*>(b2 + (8 + n) * 96)[tid];
  }
  __syncthreads();

  // ----- layer 2 + softshrink, write back in place -----
  for (int sub = 0; sub < 6; ++sub) {
    v8f aRR = {}, aII = {}, aRI = {}, aIR = {};
    for (int k0 = 0; k0 < 96; k0 += 4) {
      v2f ar  = fragA(O1r + m0 * 96, 96, k0, lane);
      v2f ai  = fragA(O1i + m0 * 96, 96, k0, lane);
      v2f brf = fragB(Wr, 96, sub * 16, k0, lane);
      v2f bif = fragB(Wi, 96, sub * 16, k0, lane);
      aRR = wmma4(ar, brf, aRR);
      aII = wmma4(ai, bif, aII);
      aRI = wmma4(ar, bif, aRI);
      aIR = wmma4(ai, brf, aIR);
    }
    int col = sub * 16 + nn;
    float cbr = bre[col], cbi = bim[col];
    for (int i = 0; i < 8; ++i) {
      int m = m0 + i + (hi << 3);
      float vr = aRR[i] - aII[i] + cbr;
      float vi = aRI[i] + aIR[i] + cbi;
      float tr = fabsf(vr) - 0.01f;
      float ti = fabsf(vi) - 0.01f;
      vr = tr > 0.f ? copysignf(tr, vr) : 0.f;
      vi = ti > 0.f ? copysignf(ti, vi) : 0.f;
      size_t off = (row0 + m) * 768 + n * 96 + col;
      R[off] = vr;
      I[off] = vi;
    }
  }
}

// ---------------------------------------------------------------------------
// Stage 5: inverse rDFT along W (Hermitian fold) + residual add.
// out[w,c] = sum_wf s(wf)/128 * (Yr*cos(t) - Yi*sin(t)),  t = 2pi*w*wf/128
// K padded 65 -> 68.  grid (6, 8, 512), block 256
// ---------------------------------------------------------------------------
__global__ void iwdft_kernel(const float* __restrict__ inR, const float* __restrict__ inI,
                             const float* __restrict__ x, float* __restrict__ out) {
  extern __shared__ float lds[];
  const int KP = 68;
  float* Ar = lds;             // 16*68
  float* Ai = Ar + 16 * KP;
  float* Br = Ai + 16 * KP;    // 68*128
  float* Bi = Br + KP * 128;
  const int tid = threadIdx.x;
  const int bh = blockIdx.z;
  const int w0 = blockIdx.y * 16;
  const int c0 = blockIdx.x * 128;
  float4* Br4 = reinterpret_cast<float4*>(Br);
  float4* Bi4 = reinterpret_cast<float4*>(Bi);
  for (int idx = tid; idx < KP * 32; idx += 256) {
    int wf = idx >> 5, c4 = idx & 31;
    float4 vr = make_float4(0.f, 0.f, 0.f, 0.f);
    float4 vi = make_float4(0.f, 0.f, 0.f, 0.f);
    if (wf < 65) {
      size_t off = ((size_t)bh * 65 + wf) * 768 + c0;
      vr = reinterpret_cast<const float4*>(inR + off)[c4];
      vi = reinterpret_cast<const float4*>(inI + off)[c4];
    }
    Br4[idx] = vr;
    Bi4[idx] = vi;
  }
  for (int idx = tid; idx < 16 * KP; idx += 256) {
    int m = idx / KP, wf = idx % KP;
    float gr = 0.f, gi = 0.f;
    if (wf < 65) {
      float sc = (wf == 0 || wf == 64) ? (1.f / 128.f) : (2.f / 128.f);
      int p = ((w0 + m) * wf) & 127;
      float s, c;
      __sincosf((TWO_PI / 128.f) * (float)p, &s, &c);
      gr = sc * c;
      gi = -sc * s;
    }
    Ar[idx] = gr;
    Ai[idx] = gi;
  }
  __syncthreads();
  const int lane = tid & 31, wv = tid >> 5;
  const int cs = wv * 16;
  v8f acc = {};
  for (int k0 = 0; k0 < KP; k0 += 4) {
    acc = wmma4(fragA(Ar, KP, k0, lane), fragB(Br, 128, cs, k0, lane), acc);
    acc = wmma4(fragA(Ai, KP, k0, lane), fragB(Bi, 128, cs, k0, lane), acc);
  }
  const int nn = lane & 15, hi = lane >> 4;
  for (int i = 0; i < 8; ++i) {
    int w = w0 + i + (hi << 3);
    size_t off = ((size_t)bh * 128 + w) * 768 + c0 + cs + nn;
    out[off] = acc[i] + x[off];   // residual
  }
}

// ---------------------------------------------------------------------------
extern "C" void kernel_launch(void* const* d_in, const int* in_sizes, int n_in,
                              void* d_out, int out_size, void* d_ws, size_t ws_size,
                              hipStream_t stream) {
  (void)in_sizes; (void)n_in; (void)out_size; (void)ws_size;
  const float* x  = (const float*)d_in[0];
  const float* w1 = (const float*)d_in[1];
  const float* b1 = (const float*)d_in[2];
  const float* w2 = (const float*)d_in[3];
  const float* b2 = (const float*)d_in[4];
  float* out = (float*)d_out;

  const size_t P = (size_t)4 * 128 * 65 * 768;   // one freq plane (floats)
  float* Ar_ = (float*)d_ws;       // ping: stage1 out / stage4 out
  float* Ai_ = Ar_ + P;
  float* Br_ = Ai_ + P;            // pong: stage2 out / mlp in-place
  float* Bi_ = Br_ + P;

  const size_t lds1 = (size_t)(2 * 16 * 128 + 128 * 128) * sizeof(float);             //  80 KB
  const size_t lds2 = (size_t)(2 * 16 * 128 + 2 * 128 * 128) * sizeof(float);         // 144 KB
  const size_t lds3 = (size_t)(4 * 128 * 96 + 2 * 96 * 96 + 2 * 96) * sizeof(float);  // 265 KB
  const size_t lds5 = (size_t)(2 * 16 * 68 + 2 * 68 * 128) * sizeof(float);           //  77 KB

  hipFuncSetAttribute(reinterpret_cast<const void*>(wdft_kernel),
                      hipFuncAttributeMaxDynamicSharedMemorySize, (int)lds1);
  hipFuncSetAttribute(reinterpret_cast<const void*>(hdft_kernel),
                      hipFuncAttributeMaxDynamicSharedMemorySize, (int)lds2);
  hipFuncSetAttribute(reinterpret_cast<const void*>(mlp_kernel),
                      hipFuncAttributeMaxDynamicSharedMemorySize, (int)lds3);
  hipFuncSetAttribute(reinterpret_cast<const void*>(iwdft_kernel),
                      hipFuncAttributeMaxDynamicSharedMemorySize, (int)lds5);

  // 1) rDFT along W:  x -> (Ar_, Ai_)
  wdft_kernel<<<dim3(6, 5, 512), 256, lds1, stream>>>(x, Ar_, Ai_);
  // 2) DFT along H (forward): (Ar_,Ai_) -> (Br_,Bi_)
  hdft_kernel<<<dim3(6, 8, 260), 256, lds2, stream>>>(Ar_, Ai_, Br_, Bi_, -1.f);
  // 3) block-diag complex MLP + softshrink, in place on (Br_,Bi_)
  mlp_kernel<<<dim3(8, 260), 256, lds3, stream>>>(Br_, Bi_, w1, b1, w2, b2);
  // 4) inverse DFT along H: (Br_,Bi_) -> (Ar_,Ai_)
  hdft_kernel<<<dim3(6, 8, 260), 256, lds2, stream>>>(Br_, Bi_, Ar_, Ai_, 1.f);
  // 5) inverse rDFT along W + residual: (Ar_,Ai_) + x -> out
  iwdft_kernel<<<dim3(6, 8, 512), 256, lds5, stream>>>(Ar_, Ai_, x, out);
}